// PatchTokenizer_39058432590507
// MI455X (gfx1250) — compile-verified
//
#include <hip/hip_runtime.h>
#include <stdint.h>

#define AS1 __attribute__((address_space(1)))
#define AS3 __attribute__((address_space(3)))

typedef int v4i __attribute__((vector_size(16)));
typedef float v4f __attribute__((ext_vector_type(4)));

#if defined(__has_builtin)
#  if __has_builtin(__builtin_amdgcn_global_load_async_to_lds_b128) && \
      __has_builtin(__builtin_amdgcn_s_wait_asynccnt)
#    define ASYNC_BUILTIN 1
#  endif
#endif

__device__ __forceinline__ void async_copy16(const float* gp, float* lp) {
#if defined(ASYNC_BUILTIN)
  __builtin_amdgcn_global_load_async_to_lds_b128(
      (AS1 v4i*)(uintptr_t)gp, (AS3 v4i*)lp, 0, 0);
#else
  unsigned lds_off = (unsigned)(uintptr_t)(AS3 void*)lp;
  asm volatile("global_load_async_to_lds_b128 %0, %1, off"
               :: "v"(lds_off), "v"(gp) : "memory");
#endif
}

__device__ __forceinline__ void async_wait0() {
#if defined(ASYNC_BUILTIN)
  __builtin_amdgcn_s_wait_asynccnt(0);
#else
  asm volatile("s_wait_asynccnt 0" ::: "memory");
#endif
}

// Streaming b128 store with non-temporal hint (output is written once, never re-read)
__device__ __forceinline__ void store_nt4(float* p, v4f v) {
  __builtin_nontemporal_store(v, (v4f*)p);
}

// One block per 32x32 tile of one image. Computes e16 (4 vals), e32, m16, m32
// and writes p16m / p32m / f32m for the tile, all from a single LDS staging
// of the 3x32x32 pixel block (HBM read exactly once, each output written once).
__global__ __launch_bounds__(256) void tok_tile_kernel(
    const float* __restrict__ images,
    const float* __restrict__ mean3,
    const float* __restrict__ std3,
    float* __restrict__ out_e16, float* __restrict__ out_e32,
    float* __restrict__ out_m16, float* __restrict__ out_m32,
    float* __restrict__ out_p16, float* __restrict__ out_p32,
    float* __restrict__ out_f32)
{
  __shared__ float pix[3][32][32];      // 12 KB raw pixels (normalized space)
  __shared__ unsigned hist[4][512];     // 8 KB: four 16x16 sub-patch histograms
  __shared__ float wred[5][8];          // per-wave partials (4 subs + tile)
  __shared__ float sval[8];             // e16[0..3], e32, m32, m16

  const int t   = threadIdx.x;
  const int blk = blockIdx.x;
  const int tj  = blk & 15;
  const int ti  = (blk >> 4) & 15;
  const int b   = blk >> 8;

  // zero histograms (2048 entries / 256 threads)
  unsigned* hflat = &hist[0][0];
  #pragma unroll
  for (int i = 0; i < 8; ++i) hflat[t + 256 * i] = 0u;

  // async-DMA stage: each thread brings one float4 per channel
  const int row = t >> 3;            // 0..31
  const int c4  = (t & 7) << 2;      // 0,4,...,28
  #pragma unroll
  for (int c = 0; c < 3; ++c) {
    const float* gp = images +
        ((((size_t)(b * 3 + c)) * 512 + (ti * 32 + row)) * 512 + (tj * 32 + c4));
    async_copy16(gp, &pix[c][row][c4]);
  }
  async_wait0();
  __syncthreads();

  // grayscale + per-subpatch histogram for this thread's 4 pixels
  const float m0 = mean3[0], m1 = mean3[1], m2 = mean3[2];
  const float s0 = std3[0],  s1 = std3[1],  s2 = std3[2];
  #pragma unroll
  for (int k = 0; k < 4; ++k) {
    const int col = c4 + k;
    float r  = fminf(fmaxf((pix[0][row][col] * s0 + m0) * 255.f, 0.f), 255.f);
    float g  = fminf(fmaxf((pix[1][row][col] * s1 + m1) * 255.f, 0.f), 255.f);
    float bl = fminf(fmaxf((pix[2][row][col] * s2 + m2) * 255.f, 0.f), 255.f);
    float gray = r * 0.2989f + g * 0.587f + bl * 0.114f;
    int bin = (int)(gray * 2.0f);                 // bins/256 == 2
    bin = bin < 0 ? 0 : (bin > 511 ? 511 : bin);
    const int sub = ((row >> 4) << 1) | (col >> 4);
    atomicAdd(&hist[sub][bin], 1u);               // ds_add_u32
  }
  __syncthreads();

  // entropies: 4 sub-patches (256 px) + full tile (1024 px = sum of hists)
  float part[5] = {0.f, 0.f, 0.f, 0.f, 0.f};
  #pragma unroll
  for (int rb = 0; rb < 2; ++rb) {
    const int bin = t + (rb << 8);
    const unsigned h0 = hist[0][bin], h1 = hist[1][bin];
    const unsigned h2 = hist[2][bin], h3 = hist[3][bin];
    const float p0 = (float)h0 * (1.f / 256.f);
    const float p1 = (float)h1 * (1.f / 256.f);
    const float p2 = (float)h2 * (1.f / 256.f);
    const float p3 = (float)h3 * (1.f / 256.f);
    const float pc = (float)(h0 + h1 + h2 + h3) * (1.f / 1024.f);
    part[0] += p0 * log2f(p0 + 1e-10f);
    part[1] += p1 * log2f(p1 + 1e-10f);
    part[2] += p2 * log2f(p2 + 1e-10f);
    part[3] += p3 * log2f(p3 + 1e-10f);
    part[4] += pc * log2f(pc + 1e-10f);
  }
  const int lane = t & 31, wave = t >> 5;         // wave32
  #pragma unroll
  for (int s = 0; s < 5; ++s) {
    float v = part[s];
    #pragma unroll
    for (int off = 16; off > 0; off >>= 1) v += __shfl_down(v, off);
    if (lane == 0) wred[s][wave] = v;
  }
  __syncthreads();
  if (t == 0) {
    #pragma unroll
    for (int s = 0; s < 5; ++s) {
      float acc = 0.f;
      #pragma unroll
      for (int wv = 0; wv < 8; ++wv) acc += wred[s][wv];
      sval[s] = -acc;
    }
    const float m32v = (sval[4] < 7.0f) ? 1.f : 0.f;
    sval[5] = m32v;          // m32
    sval[6] = 1.f - m32v;    // m16 (same for all 4 sub-patches)
  }
  __syncthreads();

  const float m32f = sval[5];
  const float m16f = sval[6];

  // scalar outputs
  if (t < 4) {
    const int si = t >> 1, sj = t & 1;
    const int h = ti * 2 + si, w = tj * 2 + sj;
    out_e16[((size_t)b * 32 + h) * 32 + w] = sval[t];
    out_m16[((size_t)b * 32 + h) * 32 + w] = m16f;
  } else if (t == 4) {
    out_e32[((size_t)b * 16 + ti) * 16 + tj] = sval[4];
    out_m32[((size_t)b * 16 + ti) * 16 + tj] = m32f;
  }

  // p16m + f32m: same LDS reads, two masked NT b128 stores each (768 float4s)
  #pragma unroll
  for (int k = 0; k < 3; ++k) {
    const int j    = t + (k << 8);       // 0..767, lane-consecutive -> coalesced
    const int sub  = j / 192;
    const int rem  = j - sub * 192;
    const int c    = rem >> 6;
    const int rem2 = rem & 63;
    const int p    = rem2 >> 2;
    const int q4   = (rem2 & 3) << 2;
    const int si = sub >> 1, sj = sub & 1;
    const v4f v = *(const v4f*)&pix[c][si * 16 + p][sj * 16 + q4];
    const int h = ti * 2 + si, w = tj * 2 + sj;
    const size_t o16 = ((((size_t)b * 32 + h) * 32 + w) * 3 + c) * 256 + p * 16 + q4;
    store_nt4(&out_p16[o16], v * m16f);
    const size_t of = (((((size_t)b * 16 + ti) * 16 + tj) * 4 + sub) * 3 + c) * 256
                      + p * 16 + q4;
    store_nt4(&out_f32[of], v * m32f);
  }

  // p32m: exact 2x2 average pool of raw images, masked (192 NT float4s)
  if (t < 192) {
    const int c   = t / 64;
    const int rem = t - c * 64;
    const int p   = rem >> 2;
    const int q4  = (rem & 3) << 2;
    const v4f r0a = *(const v4f*)&pix[c][2 * p    ][2 * q4    ];
    const v4f r0b = *(const v4f*)&pix[c][2 * p    ][2 * q4 + 4];
    const v4f r1a = *(const v4f*)&pix[c][2 * p + 1][2 * q4    ];
    const v4f r1b = *(const v4f*)&pix[c][2 * p + 1][2 * q4 + 4];
    v4f o;
    o.x = (r0a.x + r0a.y + r1a.x + r1a.y);
    o.y = (r0a.z + r0a.w + r1a.z + r1a.w);
    o.z = (r0b.x + r0b.y + r1b.x + r1b.y);
    o.w = (r0b.z + r0b.w + r1b.z + r1b.w);
    o = o * (0.25f * m32f);
    const size_t o32 = ((((size_t)b * 16 + ti) * 16 + tj) * 3 + c) * 256 + p * 16 + q4;
    store_nt4(&out_p32[o32], o);
  }
}

// seqlens[b] = sum(m16) + sum(m32) = sum over 256 tiles of (4 - 3*m32)
__global__ __launch_bounds__(256) void seqlen_kernel(const float* __restrict__ m32,
                                                     float* __restrict__ seqlens) {
  __shared__ float red[256];
  const int b = blockIdx.x, t = threadIdx.x;
  red[t] = 4.f - 3.f * m32[(size_t)b * 256 + t];
  __syncthreads();
  for (int off = 128; off > 0; off >>= 1) {
    if (t < off) red[t] += red[t + off];
    __syncthreads();
  }
  if (t == 0) seqlens[b] = red[0];
}

extern "C" void kernel_launch(void* const* d_in, const int* in_sizes, int n_in,
                              void* d_out, int out_size, void* d_ws, size_t ws_size,
                              hipStream_t stream) {
  const float* images = (const float*)d_in[0];
  const float* mean3  = (const float*)d_in[1];
  const float* std3   = (const float*)d_in[2];
  const int B = in_sizes[0] / (3 * 512 * 512);   // 16

  float* out = (float*)d_out;
  size_t off = 0;
  float* e16 = out + off; off += (size_t)B * 32 * 32;
  float* e32 = out + off; off += (size_t)B * 16 * 16;
  float* m16 = out + off; off += (size_t)B * 32 * 32;
  float* m32 = out + off; off += (size_t)B * 16 * 16;
  float* seq = out + off; off += (size_t)B;
  float* p16 = out + off; off += (size_t)B * 32 * 32 * 3 * 16 * 16;
  float* p32 = out + off; off += (size_t)B * 16 * 16 * 3 * 16 * 16;
  float* f32o = out + off;

  tok_tile_kernel<<<dim3(B * 16 * 16), dim3(256), 0, stream>>>(
      images, mean3, std3, e16, e32, m16, m32, p16, p32, f32o);
  seqlen_kernel<<<dim3(B), dim3(256), 0, stream>>>(m32, seq);
}